// FALCON_61065845015425
// MI455X (gfx1250) — compile-verified
//
#include <hip/hip_runtime.h>
#include <hip/hip_bf16.h>
#include <math.h>

typedef __attribute__((ext_vector_type(2))) float v2f;
typedef __attribute__((ext_vector_type(8))) float v8f;

#define BB 32
#define SS 512
#define HH 768
#define NT_PER_WAVE 6   // 48 N-tiles (768/16) / 8 waves

// ---------------------------------------------------------------------------
// Kernel 1: span means (direct gather, no masked einsum) + softmax entity agg
// grid = 3 sources * 32 batches; block = 256 threads (8 waves)
// agg layout: [src(3)][type(4)][B][H]
// ---------------------------------------------------------------------------
__global__ __launch_bounds__(256) void span_agg_kernel(
    const float* __restrict__ bert, const float* __restrict__ t1bert,
    const float* __restrict__ t2bert, const int* __restrict__ st,
    const int* __restrict__ en, const float* __restrict__ W_ea,
    const float* __restrict__ b_ea, float* __restrict__ agg)
{
    const int src = blockIdx.x / BB;
    const int b   = blockIdx.x % BB;
    const float* h = (src == 0) ? bert : (src == 1 ? t1bert : t2bert);
    h += (size_t)b * SS * HH;
    const int tid  = threadIdx.x;
    const int lane = tid & 31, wave = tid >> 5;
    __shared__ float wpart[4][8];
    __shared__ float wts[4];
    const float bea = b_ea[0];

    for (int t = 0; t < 4; ++t) {
        float m[4][3];  // 4 spans x 3 H-elements per thread (768/256)
        #pragma unroll
        for (int e = 0; e < 4; ++e) {
            const int idx = (b * 4 + t) * 4 + e;
            const int s0 = st[idx], e0 = en[idx];
            const int cnt = e0 - s0 - 1;                 // tokens strictly inside
            const float inv = 1.0f / (float)(cnt > 1 ? cnt : 1);
            #pragma unroll
            for (int i = 0; i < 3; ++i) {
                const int hh = tid + 256 * i;
                float s = 0.f;
                for (int p = s0 + 1; p < e0; ++p) s += h[(size_t)p * HH + hh];
                m[e][i] = s * inv;
            }
        }
        // scores: dot(mean_e, W_ea)
        float part[4] = {0.f, 0.f, 0.f, 0.f};
        #pragma unroll
        for (int i = 0; i < 3; ++i) {
            const float w = W_ea[tid + 256 * i];
            #pragma unroll
            for (int e = 0; e < 4; ++e) part[e] += m[e][i] * w;
        }
        #pragma unroll
        for (int off = 16; off > 0; off >>= 1) {
            #pragma unroll
            for (int e = 0; e < 4; ++e) part[e] += __shfl_down(part[e], off, 32);
        }
        if (lane == 0) {
            #pragma unroll
            for (int e = 0; e < 4; ++e) wpart[e][wave] = part[e];
        }
        __syncthreads();
        if (tid == 0) {
            float sc[4];
            for (int e = 0; e < 4; ++e) {
                float s = 0.f;
                for (int w = 0; w < 8; ++w) s += wpart[e][w];
                sc[e] = s + bea;
            }
            float mx = fmaxf(fmaxf(sc[0], sc[1]), fmaxf(sc[2], sc[3]));
            float ex[4], sum = 0.f;
            for (int e = 0; e < 4; ++e) { ex[e] = __expf(sc[e] - mx); sum += ex[e]; }
            for (int e = 0; e < 4; ++e) wts[e] = ex[e] / sum;
        }
        __syncthreads();
        const float w0 = wts[0], w1 = wts[1], w2 = wts[2], w3 = wts[3];
        float* dst = agg + (((size_t)src * 4 + t) * BB + b) * HH;
        #pragma unroll
        for (int i = 0; i < 3; ++i)
            dst[tid + 256 * i] = w0 * m[0][i] + w1 * m[1][i] + w2 * m[2][i] + w3 * m[3][i];
        __syncthreads();
    }
}

// ---------------------------------------------------------------------------
// Generic [16 x K] @ [K x 768] tile GEMM with V_WMMA_F32_16X16X4_F32.
// Block = 256 threads = 8 waves; wave owns 6 consecutive 16x16 N-tiles.
// X chunk (16 x 768, pre-op applied) staged in LDS; W rows read from global
// (half-wave reads a contiguous 64B row segment per operand).
// preop: 0=identity 1=tanh ; postop: 0=+bias 1=tanh(y+b) 2=sigmoid(y+b)*aux
// ---------------------------------------------------------------------------
__device__ __forceinline__ void gemm_mtile(
    const float* __restrict__ X, int ldx, int preop,
    const float* __restrict__ W, const float* __restrict__ bias,
    int K, float* __restrict__ out, int ldo, int postop,
    const float* __restrict__ aux, int m0, float* Xs)
{
    const int tid  = threadIdx.x;
    const int wave = tid >> 5, lane = tid & 31;
    const int half = lane >> 4, l15 = lane & 15;

    const v8f vzero = {0.f, 0.f, 0.f, 0.f, 0.f, 0.f, 0.f, 0.f};
    v8f acc[NT_PER_WAVE];
    #pragma unroll
    for (int j = 0; j < NT_PER_WAVE; ++j) acc[j] = vzero;

    const int n_base = wave * (NT_PER_WAVE * 16);

    for (int kc = 0; kc < K; kc += HH) {
        __syncthreads();
        for (int r = 0; r < 16; ++r) {
            const float* xr = X + (size_t)(m0 + r) * ldx + kc;
            for (int c = tid; c < HH; c += 256) {
                float v = xr[c];
                Xs[r * HH + c] = preop ? tanhf(v) : v;
            }
        }
        __syncthreads();
        // A layout (16x4 f32): lanes 0-15 row M=l15 hold K={0,1}; lanes 16-31 hold K={2,3}
        const float* xrow = Xs + l15 * HH + half * 2;
        for (int k0 = 0; k0 < HH; k0 += 4) {
            v2f a;
            a.x = xrow[k0];
            a.y = xrow[k0 + 1];
            // B layout (4x16 f32): lanes 0-15 rows K={0,1}, lanes 16-31 rows K={2,3}
            const float* wr = W + (size_t)(kc + k0 + half * 2) * HH;
            #pragma unroll
            for (int j = 0; j < NT_PER_WAVE; ++j) {
                const int n = n_base + j * 16 + l15;
                v2f bb;
                bb.x = wr[n];
                bb.y = wr[HH + n];
                acc[j] = __builtin_amdgcn_wmma_f32_16x16x4_f32(
                    false, a, false, bb, (short)0, acc[j], false, false);
            }
        }
    }
    // C/D layout: VGPR i -> lanes 0-15: M=i, N=l15 ; lanes 16-31: M=8+i
    #pragma unroll
    for (int j = 0; j < NT_PER_WAVE; ++j) {
        const int n = n_base + j * 16 + l15;
        const float bi = bias[n];
        #pragma unroll
        for (int i = 0; i < 8; ++i) {
            const int mrow = m0 + half * 8 + i;
            float y = acc[j][i] + bi;
            if (postop == 1)      y = tanhf(y);
            else if (postop == 2) y = aux[(size_t)mrow * HH + n] / (1.0f + __expf(-y));
            out[(size_t)mrow * ldo + n] = y;
        }
    }
}

// ---------------------------------------------------------------------------
// Stage 2: 15 GEMMs [32x768]@[768x768] -> main_cat / tra1 / tra2 (+ gates)
// grid = 30 blocks (gemm id * 2 M-tiles)
// ---------------------------------------------------------------------------
__global__ __launch_bounds__(256) void gemm_stage2(
    const float* pool, const float* t1pool, const float* t2pool,
    const float* t1layer, const float* t2layer,
    const float* Wcls, const float* bcls,
    const float* Wname, const float* bname,
    const float* Wtime, const float* btime,
    const float* Wplace, const float* bplace,
    const float* Wgate, const float* bgate,
    const float* agg, float* main_cat, float* tra1, float* tra2)
{
    __shared__ float Xs[16 * HH];
    const int g  = blockIdx.x >> 1;
    const int m0 = (blockIdx.x & 1) * 16;
    const float *X, *W, *bias, *aux = nullptr;
    float* out; int ldo; int preop = 1, postop = 0;
    #define AGG(s, t) (agg + (((size_t)(s) * 4 + (t)) * BB) * HH)
    switch (g) {
      case 0:  X = pool;       W = Wcls;   bias = bcls;   out = main_cat;          ldo = 7*HH; break;
      case 1:  X = AGG(0, 0);  W = Wname;  bias = bname;  out = main_cat + 1*HH;   ldo = 7*HH; break;
      case 2:  X = AGG(0, 1);  W = Wname;  bias = bname;  out = main_cat + 2*HH;   ldo = 7*HH; break;
      case 3:  X = AGG(0, 2);  W = Wtime;  bias = btime;  out = main_cat + 3*HH;   ldo = 7*HH; break;
      case 4:  X = AGG(0, 3);  W = Wplace; bias = bplace; out = main_cat + 4*HH;   ldo = 7*HH; break;
      case 5:  X = t1pool;     W = Wcls;   bias = bcls;   out = tra1;              ldo = 4*HH; break;
      case 6:  X = AGG(1, 0);  W = Wname;  bias = bname;  out = tra1 + 1*HH;       ldo = 4*HH; break;
      case 7:  X = AGG(1, 2);  W = Wtime;  bias = btime;  out = tra1 + 2*HH;       ldo = 4*HH; break;
      case 8:  X = AGG(1, 3);  W = Wplace; bias = bplace; out = tra1 + 3*HH;       ldo = 4*HH; break;
      case 9:  X = t2pool;     W = Wcls;   bias = bcls;   out = tra2;              ldo = 4*HH; break;
      case 10: X = AGG(2, 1);  W = Wname;  bias = bname;  out = tra2 + 1*HH;       ldo = 4*HH; break;
      case 11: X = AGG(2, 2);  W = Wtime;  bias = btime;  out = tra2 + 2*HH;       ldo = 4*HH; break;
      case 12: X = AGG(2, 3);  W = Wplace; bias = bplace; out = tra2 + 3*HH;       ldo = 4*HH; break;
      case 13: X = t1layer;    W = Wgate;  bias = bgate;  out = main_cat + 5*HH;   ldo = 7*HH;
               preop = 0; postop = 2; aux = t1layer; break;
      default: X = t2layer;    W = Wgate;  bias = bgate;  out = main_cat + 6*HH;   ldo = 7*HH;
               preop = 0; postop = 2; aux = t2layer; break;
    }
    #undef AGG
    gemm_mtile(X, HH, preop, W, bias, HH, out, ldo, postop, aux, m0, Xs);
}

// ---------------------------------------------------------------------------
// Stage 3: hidden = tanh(cat @ W + b) for the three heads. grid = 6 blocks.
// ---------------------------------------------------------------------------
__global__ __launch_bounds__(256) void gemm_stage3(
    const float* main_cat, const float* tra1, const float* tra2,
    const float* Wall, const float* ball,
    const float* Wall1, const float* ball1, float* hid)
{
    __shared__ float Xs[16 * HH];
    const int g  = blockIdx.x >> 1;
    const int m0 = (blockIdx.x & 1) * 16;
    if (g == 0)
        gemm_mtile(main_cat, 7*HH, 0, Wall,  ball,  7*HH, hid,             HH, 1, nullptr, m0, Xs);
    else if (g == 1)
        gemm_mtile(tra1,     4*HH, 0, Wall1, ball1, 4*HH, hid + BB*HH,     HH, 1, nullptr, m0, Xs);
    else
        gemm_mtile(tra2,     4*HH, 0, Wall1, ball1, 4*HH, hid + 2*BB*HH,   HH, 1, nullptr, m0, Xs);
}

// ---------------------------------------------------------------------------
// Stage 4: logits [3][32][2] = hid @ W{2,3} + b. One wave per output scalar.
// grid = 24 blocks * 8 waves = 192 waves.
// ---------------------------------------------------------------------------
__global__ __launch_bounds__(256) void finals_kernel(
    const float* __restrict__ hid,
    const float* __restrict__ W2, const float* __restrict__ b2,
    const float* __restrict__ W3, const float* __restrict__ b3,
    float* __restrict__ out)
{
    const int gw   = blockIdx.x * 8 + (threadIdx.x >> 5);
    const int lane = threadIdx.x & 31;
    const int s = gw / 64, r = gw % 64, b = r >> 1, o = r & 1;
    const float* h  = hid + ((size_t)s * BB + b) * HH;
    const float* Wm = (s == 0) ? W2 : W3;
    const float* bm = (s == 0) ? b2 : b3;
    float p = 0.f;
    for (int k = lane; k < HH; k += 32) p += h[k] * Wm[k * 2 + o];
    #pragma unroll
    for (int off = 16; off > 0; off >>= 1) p += __shfl_down(p, off, 32);
    if (lane == 0) out[s * 64 + b * 2 + o] = p + bm[o];
}

// ---------------------------------------------------------------------------
extern "C" void kernel_launch(void* const* d_in, const int* in_sizes, int n_in,
                              void* d_out, int out_size, void* d_ws, size_t ws_size,
                              hipStream_t stream)
{
    (void)in_sizes; (void)n_in; (void)out_size; (void)ws_size;
    const float* bert   = (const float*)d_in[0];
    const float* pool   = (const float*)d_in[1];
    const float* t1bert = (const float*)d_in[2];
    const float* t1pool = (const float*)d_in[3];
    const float* t2bert = (const float*)d_in[4];
    const float* t2pool = (const float*)d_in[5];
    const float* t1lay  = (const float*)d_in[6];
    const float* t2lay  = (const float*)d_in[7];
    const int*   st     = (const int*)d_in[8];
    const int*   en     = (const int*)d_in[9];
    const float* W_ea   = (const float*)d_in[10];
    const float* b_ea   = (const float*)d_in[11];
    const float* Wcls   = (const float*)d_in[12];
    const float* bcls   = (const float*)d_in[13];
    const float* Wname  = (const float*)d_in[14];
    const float* bname  = (const float*)d_in[15];
    const float* Wtime  = (const float*)d_in[16];
    const float* btime  = (const float*)d_in[17];
    const float* Wplace = (const float*)d_in[18];
    const float* bplace = (const float*)d_in[19];
    const float* Wall   = (const float*)d_in[20];
    const float* ball   = (const float*)d_in[21];
    const float* Wall1  = (const float*)d_in[22];
    const float* ball1  = (const float*)d_in[23];
    // d_in[24]=W_q, d_in[25]=b_q: dead (softmax over size-1 axis -> attn(x)==x)
    const float* Wall2  = (const float*)d_in[26];
    const float* ball2  = (const float*)d_in[27];
    const float* Wall3  = (const float*)d_in[28];
    const float* ball3  = (const float*)d_in[29];
    const float* Wgate  = (const float*)d_in[30];
    const float* bgate  = (const float*)d_in[31];

    float* ws       = (float*)d_ws;
    float* agg      = ws;                     // [3][4][32][768]
    float* main_cat = agg + 12 * BB * HH;     // [32][7*768]
    float* tra1     = main_cat + BB * 7 * HH; // [32][4*768]
    float* tra2     = tra1 + BB * 4 * HH;     // [32][4*768]
    float* hid      = tra2 + BB * 4 * HH;     // [3][32][768]
    float* outp     = (float*)d_out;          // 3*[32][2] = 192 floats

    span_agg_kernel<<<3 * BB, 256, 0, stream>>>(bert, t1bert, t2bert, st, en,
                                                W_ea, b_ea, agg);
    gemm_stage2<<<30, 256, 0, stream>>>(pool, t1pool, t2pool, t1lay, t2lay,
                                        Wcls, bcls, Wname, bname, Wtime, btime,
                                        Wplace, bplace, Wgate, bgate,
                                        agg, main_cat, tra1, tra2);
    gemm_stage3<<<6, 256, 0, stream>>>(main_cat, tra1, tra2, Wall, ball,
                                       Wall1, ball1, hid);
    finals_kernel<<<24, 256, 0, stream>>>(hid, Wall2, ball2, Wall3, ball3, outp);
}